// Sampler_50225347559928
// MI455X (gfx1250) — compile-verified
//
#include <hip/hip_runtime.h>
#include <stdint.h>

// ---------------------------------------------------------------------------
// MI455X sampler: temperature softmax -> top-k -> top-p -> exp-noise argmax.
//
// Roofline: 65.5 MB logits @ 23.3 TB/s => ~2.8us per streaming pass; the row
// data is L2-resident afterwards (65.5MB < 192MB L2) so re-collection passes
// are cheap. One workgroup (1024 thr = 32 wave32) per row streams via the
// CDNA5 async path: global_load_async_to_lds_b128, throttled by
// s_wait_asynccnt, double-buffered, 4 x b128 per lane per tile =>
// 8 outstanding async ops/wave -> ~128KB in flight per WGP (latency hiding
// sized for ~182 GB/s per-WGP share at ~800ns HBM latency).
//
// Selection: pass A = online softmax (M,S). Candidates with the provable
// top-p bound p >= 0.1/V are collected; if a flat row overflows the LDS
// candidate buffer, bisect the value threshold (each probe is an L2-resident
// re-stream) until count in [top_k, CAP] -- an upper value-set of size >=
// top_k contains the exact top-k, so the final sort+cumsum is exact.
// ---------------------------------------------------------------------------

#define NT     1024            // threads per block (32 wave32 waves)
#define LPT    4               // float4 async loads per lane per tile
#define TILE4  (NT * LPT)      // float4 elements per tile (4096 -> 64KB)
#define CAP    2048            // max candidates kept per row in LDS
#define MAXTRY 10
#define TOPP   0.9f

// monotonic bijection float -> uint32 (descending-sortable), and inverse
__device__ __forceinline__ uint32_t fkey(float f) {
  uint32_t u = __float_as_uint(f);
  return (u & 0x80000000u) ? ~u : (u | 0x80000000u);
}
__device__ __forceinline__ float funkey(uint32_t k) {
  uint32_t u = (k & 0x80000000u) ? (k & 0x7fffffffu) : ~k;
  return __uint_as_float(u);
}

// deterministic Exp(1) noise clamped to >=1e-10 (splitmix64 -> -log(u))
__device__ __forceinline__ float exp_noise(uint64_t x) {
  x += 0x9E3779B97F4A7C15ull;
  x = (x ^ (x >> 30)) * 0xBF58476D1CE4E5B9ull;
  x = (x ^ (x >> 27)) * 0x94D049BB133111EBull;
  x ^= (x >> 31);
  float u = (float)((uint32_t)(x >> 40) + 1u) * (1.0f / 16777216.0f); // (0,1]
  float e = -__logf(u);
  return fmaxf(e, 1e-10f);
}

__global__ __launch_bounds__(NT) void sampler_topk_topp_kernel(
    const float* __restrict__ logits, const float* __restrict__ temps,
    const int* __restrict__ topk_ptr, int* __restrict__ out_tok, int V) {
  const int row  = blockIdx.x;
  const int tid  = threadIdx.x;
  const int V4   = V >> 2;                         // V is a multiple of 4
  const int nIt  = (V4 + TILE4 - 1) / TILE4;
  const float invT = 1.0f / temps[row];
  const unsigned long long gbase =
      (unsigned long long)(uintptr_t)(logits + (size_t)row * (size_t)V);

  __shared__ float4 buf[2][TILE4];                 // 128 KB async staging
  __shared__ unsigned long long skey[CAP];         // 16 KB candidate keys
  __shared__ float redM[NT / 32], redS[NT / 32];
  __shared__ float sM, sS, sThr, sDLo, sDHi;
  __shared__ int candCount, sDone, sKK;

  const uint32_t ldsB0 = (uint32_t)(uintptr_t)&buf[0][tid];
  const uint32_t ldsB1 = (uint32_t)(uintptr_t)&buf[1][tid];

  // ------------------- Pass A: online softmax (M, S) ----------------------
  float lm = -3.0e38f, ls = 0.0f;
  {
#pragma unroll
    for (int k = 0; k < LPT; ++k) {                // prologue: tile 0
      uint32_t off = (uint32_t)min(k * NT + tid, V4 - 1) * 16u;
      uint32_t la  = ldsB0 + (uint32_t)(k * NT * 16);
      asm volatile("global_load_async_to_lds_b128 %0, %1, %2"
                   :: "v"(la), "v"(off), "s"(gbase) : "memory");
    }
    for (int it = 0; it < nIt; ++it) {
      if (it + 1 < nIt) {
        uint32_t lb = ((it & 1) == 0) ? ldsB1 : ldsB0;
#pragma unroll
        for (int k = 0; k < LPT; ++k) {
          uint32_t off =
              (uint32_t)min((it + 1) * TILE4 + k * NT + tid, V4 - 1) * 16u;
          uint32_t la = lb + (uint32_t)(k * NT * 16);
          asm volatile("global_load_async_to_lds_b128 %0, %1, %2"
                       :: "v"(la), "v"(off), "s"(gbase) : "memory");
        }
        asm volatile("s_wait_asynccnt 4" ::: "memory");  // current tile done
      } else {
        asm volatile("s_wait_asynccnt 0" ::: "memory");
      }
      const int b = it & 1;
#pragma unroll
      for (int k = 0; k < LPT; ++k) {
        float4 v = buf[b][k * NT + tid];           // each lane eats its own 16B
        if (it * TILE4 + k * NT + tid < V4) {
          float xs[4] = {v.x * invT, v.y * invT, v.z * invT, v.w * invT};
#pragma unroll
          for (int j = 0; j < 4; ++j) {
            float x = xs[j];
            if (x <= lm) { ls += __expf(x - lm); }
            else         { ls = ls * __expf(lm - x) + 1.0f; lm = x; }
          }
        }
      }
    }
  }
  // wave32 reduction of (m, s), then cross-wave combine
#pragma unroll
  for (int o = 16; o > 0; o >>= 1) {
    float om = __shfl_xor(lm, o, 32);
    float os = __shfl_xor(ls, o, 32);
    float nm = fmaxf(lm, om);
    ls = ls * __expf(lm - nm) + os * __expf(om - nm);
    lm = nm;
  }
  if ((tid & 31) == 0) { redM[tid >> 5] = lm; redS[tid >> 5] = ls; }
  __syncthreads();
  if (tid == 0) {
    float M = redM[0], S = redS[0];
    for (int w = 1; w < NT / 32; ++w) {
      float om = redM[w], os = redS[w];
      float nm = fmaxf(M, om);
      S = S * __expf(M - nm) + os * __expf(om - nm);
      M = nm;
    }
    sM = M; sS = S;
    // top-p bound: any survivor beyond sorted pos 0 has p >= 0.1/V,
    // i.e. x >= M + ln(0.1*S/V); the argmax element satisfies it too.
    sThr = M + __logf(0.1f * S / (float)V);
    sKK  = *topk_ptr;
    sDone = 0;
  }
  __syncthreads();

  // --------- Pass B (with threshold bisection on overflow) ----------------
  const int kkReq = (sKK > 0) ? min(sKK, CAP) : 1;
  for (int attempt = 0; attempt < MAXTRY; ++attempt) {
    if (tid == 0) candCount = 0;
    __syncthreads();
    const float thr = sThr;
    {
#pragma unroll
      for (int k = 0; k < LPT; ++k) {              // prologue: tile 0
        uint32_t off = (uint32_t)min(k * NT + tid, V4 - 1) * 16u;
        uint32_t la  = ldsB0 + (uint32_t)(k * NT * 16);
        asm volatile("global_load_async_to_lds_b128 %0, %1, %2"
                     :: "v"(la), "v"(off), "s"(gbase) : "memory");
      }
      for (int it = 0; it < nIt; ++it) {
        if (it + 1 < nIt) {
          uint32_t lb = ((it & 1) == 0) ? ldsB1 : ldsB0;
#pragma unroll
          for (int k = 0; k < LPT; ++k) {
            uint32_t off =
                (uint32_t)min((it + 1) * TILE4 + k * NT + tid, V4 - 1) * 16u;
            uint32_t la = lb + (uint32_t)(k * NT * 16);
            asm volatile("global_load_async_to_lds_b128 %0, %1, %2"
                         :: "v"(la), "v"(off), "s"(gbase) : "memory");
          }
          asm volatile("s_wait_asynccnt 4" ::: "memory");
        } else {
          asm volatile("s_wait_asynccnt 0" ::: "memory");
        }
        const int b = it & 1;
#pragma unroll
        for (int k = 0; k < LPT; ++k) {
          float4 v = buf[b][k * NT + tid];
          int e4 = it * TILE4 + k * NT + tid;
          if (e4 < V4) {
            float xs[4] = {v.x * invT, v.y * invT, v.z * invT, v.w * invT};
#pragma unroll
            for (int j = 0; j < 4; ++j) {
              if (xs[j] >= thr) {
                int p = atomicAdd(&candCount, 1);
                if (p < CAP) {
                  uint32_t idx = (uint32_t)(e4 * 4 + j);
                  skey[p] = ((unsigned long long)fkey(xs[j]) << 32) |
                            (unsigned long long)(~idx);  // low idx wins ties
                }
              }
            }
          }
        }
      }
    }
    __syncthreads();
    if (tid == 0) {
      int cnt = candCount;
      if (attempt == 0) {
        if (cnt <= CAP) sDone = 1;                 // safe top-p threshold fit
        else { sDLo = 0.0f; sDHi = sM - sThr; sThr = sM - 0.5f * sDHi; }
      } else {
        if (cnt > CAP)        sDHi = sM - sThr;    // too many: raise thr
        else if (cnt < kkReq) sDLo = sM - sThr;    // too few for top-k: lower
        else                  sDone = 1;           // count in [top_k, CAP]
        if (!sDone) sThr = sM - 0.5f * (sDLo + sDHi);
      }
    }
    __syncthreads();
    if (sDone) break;
  }

  // ------------------- sort candidates descending (bitonic) ---------------
  const int n = min(candCount, CAP);
  int nP = 1;
  while (nP < n) nP <<= 1;
  for (int i = tid; i < nP; i += NT)
    if (i >= n) skey[i] = 0ull;                    // smallest possible key
  __syncthreads();
  for (int k = 2; k <= nP; k <<= 1) {
    for (int j = k >> 1; j > 0; j >>= 1) {
      for (int i = tid; i < nP; i += NT) {
        int ixj = i ^ j;
        if (ixj > i) {
          unsigned long long a = skey[i], b = skey[ixj];
          bool desc = ((i & k) == 0);
          if (desc ? (a < b) : (a > b)) { skey[i] = b; skey[ixj] = a; }
        }
      }
      __syncthreads();
    }
  }

  // ------------- top-k cut, top-p cumsum, exp-noise argmax ----------------
  if (tid == 0) {
    const float M = sM, invS = 1.0f / sS;
    int kk = sKK;
    int K = n;
    if (kk > 0 && kk < K) K = kk;                  // top-k keeps first K sorted
    float cum = 0.0f, best = -3.0e38f;
    int bestIdx = 0;
    for (int i = 0; i < K; ++i) {
      unsigned long long kv = skey[i];
      float x = funkey((uint32_t)(kv >> 32));
      int idx = (int)(~(uint32_t)kv);
      float p = __expf(x - M) * invS;
      cum += p;
      if (i > 0 && cum > TOPP) break;              // tail removed (first kept)
      float score = p / exp_noise(((uint64_t)row << 32) | (uint32_t)idx);
      if (score > best) { best = score; bestIdx = idx; }
    }
    out_tok[row] = bestIdx;
  }
}

extern "C" void kernel_launch(void* const* d_in, const int* in_sizes, int n_in,
                              void* d_out, int out_size, void* d_ws, size_t ws_size,
                              hipStream_t stream) {
  const float* logits = (const float*)d_in[0];
  const float* temps  = (const float*)d_in[1];
  const int*   topk   = (const int*)d_in[2];
  const int B = in_sizes[1];
  const int V = in_sizes[0] / B;
  sampler_topk_topp_kernel<<<dim3(B), dim3(NT), 0, stream>>>(
      logits, temps, topk, (int*)d_out, V);
}